// FocalLoss_77670188580872
// MI455X (gfx1250) — compile-verified
//
#include <hip/hip_runtime.h>
#include <math.h>

#define BLK 256
#define DEG2RAD 0.017453292519943295f

// CDNA5 async global->LDS staging path (guarded; falls back to plain loads).
#if defined(__has_builtin)
#if __has_builtin(__builtin_amdgcn_global_load_async_to_lds_b128) && \
    __has_builtin(__builtin_amdgcn_global_load_async_to_lds_b32)
#define USE_ASYNC_LDS 1
#endif
#endif
#ifndef USE_ASYNC_LDS
#define USE_ASYNC_LDS 0
#endif

#if USE_ASYNC_LDS
typedef int i32x4 __attribute__((vector_size(16)));
typedef __attribute__((address_space(1))) i32x4* g_v4p;
typedef __attribute__((address_space(3))) i32x4* l_v4p;
typedef __attribute__((address_space(1))) int*   g_i32p;
typedef __attribute__((address_space(3))) int*   l_i32p;
#endif

// ---------------------------------------------------------------------------
// Kernel 1: per-anchor assignment + focal + smooth-L1, block partial sums.
// grid = (ceil(A/BLK), B), block = BLK.
// Dynamic LDS layout: ann[M*6] | poly[M*8] | rect[M*4] | areab[M] | red[3*BLK]
// ---------------------------------------------------------------------------
__global__ void focal_main_kernel(const float* __restrict__ cls,
                                  const float* __restrict__ reg,
                                  const float* __restrict__ anc,
                                  const float* __restrict__ ann,
                                  float* __restrict__ partials,
                                  int A, int M, int C, int nblk)
{
    extern __shared__ float sm[];
    float* s_ann  = sm;              // M*6
    float* s_poly = s_ann + M * 6;   // M*8  (x0,y0,...,x3,y3)
    float* s_rect = s_poly + M * 8;  // M*4  (xmin,ymin,xmax,ymax)
    float* s_ab   = s_rect + M * 4;  // M    (hrect area)
    float* s_red  = s_ab + M;        // 3*BLK

    const int tid = threadIdx.x;
    const int img = blockIdx.y;

    // --- Stage raw annotations into LDS -----------------------------------
    {
        const float* src = ann + (size_t)img * M * 6;
        const int totalF = M * 6;
#if USE_ASYNC_LDS
        const int nb128 = totalF >> 2;   // 16B chunks
        for (int i = tid; i < nb128; i += BLK) {
            __builtin_amdgcn_global_load_async_to_lds_b128(
                (g_v4p)(src + i * 4), (l_v4p)(s_ann + i * 4), 0, 0);
        }
        for (int i = (nb128 << 2) + tid; i < totalF; i += BLK) {
            __builtin_amdgcn_global_load_async_to_lds_b32(
                (g_i32p)(src + i), (l_i32p)(s_ann + i), 0, 0);
        }
        asm volatile("s_wait_asynccnt 0x0" ::: "memory");
#else
        for (int i = tid; i < totalF; i += BLK) s_ann[i] = src[i];
#endif
    }
    __syncthreads();

    // --- Build GT geometry tables in LDS ----------------------------------
    for (int j = tid; j < M; j += BLK) {
        float gcx = s_ann[j * 6 + 0], gcy = s_ann[j * 6 + 1];
        float gw  = s_ann[j * 6 + 2], gh  = s_ann[j * 6 + 3];
        float gth = s_ann[j * 6 + 4];

        float angr = gth * DEG2RAD;
        float b = cosf(angr) * 0.5f;
        float a = sinf(angr) * 0.5f;
        float p0x = gcx - a * gh - b * gw, p0y = gcy + b * gh - a * gw;
        float p1x = gcx + a * gh - b * gw, p1y = gcy - b * gh - a * gw;
        float p2x = 2.f * gcx - p0x,       p2y = 2.f * gcy - p0y;
        float p3x = 2.f * gcx - p1x,       p3y = 2.f * gcy - p1y;
        s_poly[j * 8 + 0] = p0x; s_poly[j * 8 + 1] = p0y;
        s_poly[j * 8 + 2] = p1x; s_poly[j * 8 + 3] = p1y;
        s_poly[j * 8 + 4] = p2x; s_poly[j * 8 + 5] = p2y;
        s_poly[j * 8 + 6] = p3x; s_poly[j * 8 + 7] = p3y;

        float xmn = fminf(fminf(p0x, p1x), fminf(p2x, p3x));
        float xmx = fmaxf(fmaxf(p0x, p1x), fmaxf(p2x, p3x));
        float ymn = fminf(fminf(p0y, p1y), fminf(p2y, p3y));
        float ymx = fmaxf(fmaxf(p0y, p1y), fmaxf(p2y, p3y));
        s_rect[j * 4 + 0] = xmn; s_rect[j * 4 + 1] = ymn;
        s_rect[j * 4 + 2] = xmx; s_rect[j * 4 + 3] = ymx;
        s_ab[j] = (xmx - xmn) * (ymx - ymn);
    }
    __syncthreads();

    float cls_acc = 0.0f, reg_acc = 0.0f, pos_acc = 0.0f;
    const int a_idx = blockIdx.x * BLK + tid;

    if (a_idx < A) {
        const float* ap = anc + (size_t)a_idx * 5;
        float ax1 = ap[0], ay1 = ap[1], ax2 = ap[2], ay2 = ap[3], ath = ap[4];
        float area_a = (ax2 - ax1) * (ay2 - ay1);

        // --- hor_iou argmax over M GTs (first-max wins, like jnp.argmax) --
        float best = -1.0f; int arg = 0;
        for (int j = 0; j < M; ++j) {
            float bx1 = s_rect[j * 4 + 0], by1 = s_rect[j * 4 + 1];
            float bx2 = s_rect[j * 4 + 2], by2 = s_rect[j * 4 + 3];
            float iw = fmaxf(fminf(ax2, bx2) - fmaxf(ax1, bx1), 0.0f);
            float ih = fmaxf(fminf(ay2, by2) - fmaxf(ay1, by1), 0.0f);
            float inter = iw * ih;
            float ua = fmaxf(area_a + s_ab[j] - inter, 1e-8f);
            float iou = inter / ua;
            if (iou > best) { best = iou; arg = j; }
        }
        bool hor_pos = best >= 0.4f;
        bool pos = false;

        if (hor_pos) {
            // Prefetch only the rows we'll actually consume; the clipping
            // work below hides the latency (gfx1250 global_prefetch_b8).
            __builtin_prefetch(cls + ((size_t)img * A + a_idx) * C, 0, 1);
            __builtin_prefetch(reg + ((size_t)img * A + a_idx) * 5, 0, 1);

            // --- skew IoU (Sutherland-Hodgman, reference numerics) --------
            float px[12], py[12], qx[12], qy[12];
            int n = 4;
            for (int i = 0; i < 4; ++i) {
                px[i] = s_poly[arg * 8 + 2 * i];
                py[i] = s_poly[arg * 8 + 2 * i + 1];
            }
            float qa = 0.0f;
            for (int i = 0; i < 4; ++i) {
                int jn = (i + 1) & 3;
                qa += px[i] * py[jn] - px[jn] * py[i];
            }
            float quad_area = 0.5f * fabsf(qa);

            const float pl[4][3] = { { 1.f, 0.f, -ax1}, {-1.f, 0.f, ax2},
                                     { 0.f, 1.f, -ay1}, { 0.f,-1.f, ay2} };
            for (int k = 0; k < 4; ++k) {
                float aa = pl[k][0], bb = pl[k][1], cc = pl[k][2];
                int m = 0;
                for (int i = 0; i < n; ++i) {
                    int jn = (i + 1 < n) ? i + 1 : 0;
                    float dp = aa * px[i]  + bb * py[i]  + cc;
                    float dq = aa * px[jn] + bb * py[jn] + cc;
                    bool inp = dp >= 0.0f, inq = dq >= 0.0f;
                    if (inp && m < 12) { qx[m] = px[i]; qy[m] = py[i]; ++m; }
                    if ((inp != inq) && m < 12) {
                        float den = dp - dq;
                        if (fabsf(den) < 1e-9f) den = 1e-9f;
                        float t = dp / den;
                        qx[m] = px[i] + t * (px[jn] - px[i]);
                        qy[m] = py[i] + t * (py[jn] - py[i]);
                        ++m;
                    }
                }
                n = m;
                for (int i = 0; i < n; ++i) { px[i] = qx[i]; py[i] = qy[i]; }
            }
            float cr = 0.0f;
            for (int i = 0; i < n; ++i) {
                int jn = (i + 1 < n) ? i + 1 : 0;
                cr += px[i] * py[jn] - px[jn] * py[i];
            }
            float inter_area = 0.5f * fabsf(cr);
            float uni = fmaxf(quad_area + area_a - inter_area, 1e-8f);
            float siou = inter_area / uni;
            pos = siou >= 0.2f;
        }

        // --- focal loss (targets==-1 when !hor_pos contributes 0) ---------
        if (hor_pos) {
            int cls_idx = (int)s_ann[arg * 6 + 5];
            const float* cp = cls + ((size_t)img * A + a_idx) * C;
            if ((C & 3) == 0) {
                const float4* cp4 = (const float4*)cp;
                for (int c4 = 0; c4 < (C >> 2); ++c4) {
                    float4 v4 = cp4[c4];
                    float vv[4] = { v4.x, v4.y, v4.z, v4.w };
                    #pragma unroll
                    for (int k = 0; k < 4; ++k) {
                        int c = c4 * 4 + k;
                        float v = fminf(fmaxf(vv[k], 1e-4f), 1.0f - 1e-4f);
                        bool t1 = pos && (c == cls_idx);
                        float af  = t1 ? 0.25f : 0.75f;
                        float pt  = t1 ? (1.0f - v) : v;
                        float bce = t1 ? (-logf(v)) : (-logf(1.0f - v));
                        cls_acc += af * pt * pt * bce;
                    }
                }
            } else {
                for (int c = 0; c < C; ++c) {
                    float v = fminf(fmaxf(cp[c], 1e-4f), 1.0f - 1e-4f);
                    bool t1 = pos && (c == cls_idx);
                    float af  = t1 ? 0.25f : 0.75f;
                    float pt  = t1 ? (1.0f - v) : v;
                    float bce = t1 ? (-logf(v)) : (-logf(1.0f - v));
                    cls_acc += af * pt * pt * bce;
                }
            }
        }

        // --- smooth-L1 regression (only pos anchors count) ----------------
        if (pos) {
            pos_acc = 1.0f;
            float aw = ax2 - ax1, ah = ay2 - ay1;
            float acx = ax1 + 0.5f * aw, acy = ay1 + 0.5f * ah;
            float gx = s_ann[arg * 6 + 0], gy = s_ann[arg * 6 + 1];
            float gw = fmaxf(s_ann[arg * 6 + 2], 1.0f);
            float gh = fmaxf(s_ann[arg * 6 + 3], 1.0f);
            float gt_th = s_ann[arg * 6 + 4];
            float rt[5];
            rt[0] = (gy - acy) / ah;
            rt[1] = (gx - acx) / aw;
            rt[2] = logf(gh / ah);
            rt[3] = logf(gw / aw);
            rt[4] = (gt_th - ath) * DEG2RAD;
            const float* rp = reg + ((size_t)img * A + a_idx) * 5;
            for (int i = 0; i < 5; ++i) {
                float d = fabsf(rt[i] - rp[i]);
                reg_acc += (d <= (1.0f / 9.0f)) ? 4.5f * d * d : (d - 0.5f / 9.0f);
            }
        }
    }

    // --- deterministic block tree reduction -------------------------------
    s_red[tid]           = cls_acc;
    s_red[BLK + tid]     = reg_acc;
    s_red[2 * BLK + tid] = pos_acc;
    __syncthreads();
    for (int s = BLK / 2; s > 0; s >>= 1) {
        if (tid < s) {
            s_red[tid]           += s_red[tid + s];
            s_red[BLK + tid]     += s_red[BLK + tid + s];
            s_red[2 * BLK + tid] += s_red[2 * BLK + tid + s];
        }
        __syncthreads();
    }
    if (tid == 0) {
        size_t o = ((size_t)img * nblk + blockIdx.x) * 3;
        partials[o + 0] = s_red[0];
        partials[o + 1] = s_red[BLK];
        partials[o + 2] = s_red[2 * BLK];
    }
}

// ---------------------------------------------------------------------------
// Kernel 2: per-image deterministic reduction + normalization.
// ---------------------------------------------------------------------------
__global__ void focal_reduce_img_kernel(const float* __restrict__ partials,
                                        float* __restrict__ imgres, int nblk)
{
    __shared__ float red[3][BLK];
    const int img = blockIdx.x;
    const int tid = threadIdx.x;
    float c = 0.f, r = 0.f, p = 0.f;
    for (int i = tid; i < nblk; i += BLK) {
        size_t o = ((size_t)img * nblk + i) * 3;
        c += partials[o + 0];
        r += partials[o + 1];
        p += partials[o + 2];
    }
    red[0][tid] = c; red[1][tid] = r; red[2][tid] = p;
    __syncthreads();
    for (int s = BLK / 2; s > 0; s >>= 1) {
        if (tid < s) {
            red[0][tid] += red[0][tid + s];
            red[1][tid] += red[1][tid + s];
            red[2][tid] += red[2][tid + s];
        }
        __syncthreads();
    }
    if (tid == 0) {
        float np = red[2][0];
        imgres[img * 2 + 0] = red[0][0] / fmaxf(np, 1.0f);
        imgres[img * 2 + 1] = red[1][0] / fmaxf(np * 5.0f, 1.0f);
    }
}

// ---------------------------------------------------------------------------
// Kernel 3: mean over images -> d_out[0..1].
// ---------------------------------------------------------------------------
__global__ void focal_final_kernel(const float* __restrict__ imgres,
                                   float* __restrict__ out, int B)
{
    if (threadIdx.x == 0 && blockIdx.x == 0) {
        float c = 0.f, r = 0.f;
        for (int i = 0; i < B; ++i) { c += imgres[i * 2]; r += imgres[i * 2 + 1]; }
        out[0] = c / (float)B;
        out[1] = r / (float)B;
    }
}

extern "C" void kernel_launch(void* const* d_in, const int* in_sizes, int n_in,
                              void* d_out, int out_size, void* d_ws, size_t ws_size,
                              hipStream_t stream) {
    const float* cls = (const float*)d_in[0];  // (B, A, C)
    const float* reg = (const float*)d_in[1];  // (B, A, 5)
    const float* anc = (const float*)d_in[2];  // (1, A, 5)
    const float* ann = (const float*)d_in[3];  // (B, M, 6)

    const int A = in_sizes[2] / 5;
    const int B = in_sizes[1] / (A * 5);
    const int C = in_sizes[0] / (B * A);
    const int M = in_sizes[3] / (B * 6);
    const int nblk = (A + BLK - 1) / BLK;

    float* partials = (float*)d_ws;                          // B*nblk*3 floats
    float* imgres   = partials + (size_t)B * nblk * 3;       // B*2 floats

    size_t smem = ((size_t)M * 19 + 3 * BLK) * sizeof(float);
    focal_main_kernel<<<dim3(nblk, B), dim3(BLK), smem, stream>>>(
        cls, reg, anc, ann, partials, A, M, C, nblk);
    focal_reduce_img_kernel<<<dim3(B), dim3(BLK), 0, stream>>>(partials, imgres, nblk);
    focal_final_kernel<<<dim3(1), dim3(32), 0, stream>>>(imgres, (float*)d_out, B);
}